// SSIMLoss_65979287601270
// MI455X (gfx1250) — compile-verified
//
#include <hip/hip_runtime.h>
#include <hip/hip_bf16.h>
#include <math.h>

typedef __attribute__((ext_vector_type(2))) float v2f;
typedef __attribute__((ext_vector_type(8))) float v8f;

#define H_IMG 256
#define W_IMG 256
#define B_IMG 128
#define TS 16            // output tile is 16x16
#define PR 32            // patch rows allocated (26 valid + zero pad)
#define PC 28            // patch cols allocated (26 valid + 2 zero pad -> K=28)
#define HRP 36           // H intermediate: padded row stride (32 rows used),
                         // transposed layout Ht[col][row]; 36 keeps 16B align
                         // and spreads 16 columns over 16 LDS banks
#define WAVES 4          // waves (tiles) per block, wave32
#define WAVE_LDS (PR*PC*2 + TS*HRP)   // floats per wave = 1792 + 576 = 2368

// ---------------------------------------------------------------------------
// Main kernel: one wave = one 16x16 tile. 105 v_wmma_f32_16x16x4_f32 per tile
// implement the 5 Gaussian convolutions as banded-matrix GEMMs.
// ---------------------------------------------------------------------------
__global__ __launch_bounds__(WAVES*32) void ssim_main(
    const float* __restrict__ img1, const float* __restrict__ img2,
    float* __restrict__ partial)
{
    __shared__ float smem[WAVES * WAVE_LDS];
    __shared__ float w_sh[11];
    __shared__ float red[WAVES * 32];

    const int tid  = threadIdx.x;
    const int lane = tid & 31;
    const int wave = tid >> 5;
    const bool lo  = lane < 16;
    const int  rl  = lo ? lane : lane - 16;   // row (A) / column (B,C,D) index

    // ---- Gaussian 1D weights (normalized), sigma = 1.5, width 11 ----
    if (tid < 11) {
        float s = 0.f;
        for (int i = 0; i < 11; ++i) {
            float d = (float)(i - 5);
            s += expf(-d * d / 4.5f);
        }
        float d = (float)(tid - 5);
        w_sh[tid] = expf(-d * d / 4.5f) / s;
    }
    __syncthreads();

    // ---- Banded weight operand, shared by both passes (see round-0 notes):
    // identical per-lane value w[c0-rl] for the H-pass B operand and the
    // V-pass A operand under the f32 WMMA register layouts.
    v2f wband[7];
    #pragma unroll
    for (int k = 0; k < 7; ++k) {
        int c0 = 4 * k + (lo ? 0 : 2);
        int d0 = c0 - rl;
        int d1 = c0 + 1 - rl;
        wband[k].x = (d0 >= 0 && d0 <= 10) ? w_sh[d0] : 0.f;
        wband[k].y = (d1 >= 0 && d1 <= 10) ? w_sh[d1] : 0.f;
    }

    // ---- Tile coordinates ----
    const int tile = blockIdx.x * WAVES + wave;        // 0 .. 32767
    const int b    = tile >> 8;
    const int ty   = (tile >> 4) & 15;
    const int tx   = tile & 15;
    const int y0   = ty * TS;
    const int x0   = tx * TS;

    float* P1 = smem + wave * WAVE_LDS;
    float* P2 = P1 + PR * PC;
    float* Ht = P2 + PR * PC;        // transposed H: Ht[col*HRP + row]

    // ---- Load 26x26 halo patches (zero-padded to 32x28) ----
    const float* base1 = img1 + (size_t)b * (H_IMG * W_IMG);
    const float* base2 = img2 + (size_t)b * (H_IMG * W_IMG);
    for (int s = lane; s < PR * PC; s += 32) {
        int r  = s / PC;
        int c  = s - r * PC;
        int iy = y0 + r - 5;
        int ix = x0 + c - 5;
        float v1 = 0.f, v2 = 0.f;
        if (r < 26 && c < 26 && iy >= 0 && iy < H_IMG && ix >= 0 && ix < W_IMG) {
            int idx = iy * W_IMG + ix;
            v1 = base1[idx];
            v2 = base2[idx];
        }
        P1[s] = v1;
        P2[s] = v2;
    }
    __syncthreads();

    // ---- Hoist all H-pass A-operand patch pairs into registers (read once,
    //      reused by all 5 fields): 2 m-groups x 7 k-steps x 2 images.
    v2f p1r[14], p2r[14];
    #pragma unroll
    for (int mg = 0; mg < 2; ++mg) {
        #pragma unroll
        for (int k = 0; k < 7; ++k) {
            int row = mg * 16 + rl;
            int c0  = 4 * k + (lo ? 0 : 2);           // even -> 8B aligned
            p1r[mg * 7 + k] = *(const v2f*)(P1 + row * PC + c0);
            p2r[mg * 7 + k] = *(const v2f*)(P2 + row * PC + c0);
        }
    }

    // ---- 5 fields: x, y, x^2, y^2, x*y ----
    v8f outF[5];
    #pragma unroll
    for (int f = 0; f < 5; ++f) {
        // ------- Horizontal pass: H = P * Bband, two 16-row M-groups -------
        v8f hacc[2];
        #pragma unroll
        for (int mg = 0; mg < 2; ++mg) {
            v8f acc = {0.f,0.f,0.f,0.f,0.f,0.f,0.f,0.f};
            #pragma unroll
            for (int k = 0; k < 7; ++k) {
                v2f pa = p1r[mg * 7 + k];
                v2f pb = p2r[mg * 7 + k];
                v2f a;
                if (f == 0)      a = pa;
                else if (f == 1) a = pb;
                else if (f == 2) a = pa * pa;
                else if (f == 3) a = pb * pb;
                else             a = pa * pb;
                acc = __builtin_amdgcn_wmma_f32_16x16x4_f32(
                    false, a, false, wband[k], (short)0, acc, false, false);
            }
            hacc[mg] = acc;
        }

        // Store H transposed: lane owns column rl, 8 consecutive rows per
        // accumulator half -> 4x ds_store_b128. Rows >=26 are exact zeros.
        {
            float* hb = Ht + rl * HRP + (lo ? 0 : 8);
            *(float4*)(hb +  0) = make_float4(hacc[0][0], hacc[0][1], hacc[0][2], hacc[0][3]);
            *(float4*)(hb +  4) = make_float4(hacc[0][4], hacc[0][5], hacc[0][6], hacc[0][7]);
            *(float4*)(hb + 16) = make_float4(hacc[1][0], hacc[1][1], hacc[1][2], hacc[1][3]);
            *(float4*)(hb + 20) = make_float4(hacc[1][4], hacc[1][5], hacc[1][6], hacc[1][7]);
        }
        __syncthreads();

        // ------- Vertical pass: Out = Wband * H -------
        v8f acc2 = {0.f,0.f,0.f,0.f,0.f,0.f,0.f,0.f};
        #pragma unroll
        for (int k = 0; k < 7; ++k) {
            int j0 = 4 * k + (lo ? 0 : 2);            // even -> 8B aligned
            v2f bv = *(const v2f*)(Ht + rl * HRP + j0);
            acc2 = __builtin_amdgcn_wmma_f32_16x16x4_f32(
                false, wband[k], false, bv, (short)0, acc2, false, false);
        }
        outF[f] = acc2;
        __syncthreads();   // Ht is rewritten next field
    }

    // ---- Per-pixel SSIM map + tile partial sum ----
    const float C1v = 1e-4f;   // 0.01^2
    const float C2v = 9e-4f;   // 0.03^2
    float lsum = 0.f;
    #pragma unroll
    for (int v = 0; v < 8; ++v) {
        float mu1  = outF[0][v];
        float mu2  = outF[1][v];
        float e11  = outF[2][v];
        float e22  = outF[3][v];
        float e12  = outF[4][v];
        float mu1s = mu1 * mu1;
        float mu2s = mu2 * mu2;
        float mu12 = mu1 * mu2;
        float s11  = e11 - mu1s;
        float s22  = e22 - mu2s;
        float s12  = e12 - mu12;
        float num  = (2.f * mu12 + C1v) * (2.f * s12 + C2v);
        float den  = (mu1s + mu2s + C1v) * (s11 + s22 + C2v);
        lsum += num / den;
    }

    // ---- Deterministic block reduction -> one partial per block ----
    red[tid] = lsum;
    __syncthreads();
    #pragma unroll
    for (int off = (WAVES * 32) / 2; off > 0; off >>= 1) {
        if (tid < off) red[tid] += red[tid + off];
        __syncthreads();
    }
    if (tid == 0) partial[blockIdx.x] = red[0];
}

// ---------------------------------------------------------------------------
// Fixed-order final reduction (deterministic; no float atomics).
// ---------------------------------------------------------------------------
__global__ __launch_bounds__(256) void ssim_finalize(
    const float* __restrict__ partial, int n, float* __restrict__ out)
{
    __shared__ float red[256];
    int tid = threadIdx.x;
    float s = 0.f;
    for (int i = tid; i < n; i += 256) s += partial[i];
    red[tid] = s;
    __syncthreads();
    #pragma unroll
    for (int off = 128; off > 0; off >>= 1) {
        if (tid < off) red[tid] += red[tid + off];
        __syncthreads();
    }
    if (tid == 0) out[0] = 1.0f - red[0] / 8388608.0f;   // 128*256*256
}

extern "C" void kernel_launch(void* const* d_in, const int* in_sizes, int n_in,
                              void* d_out, int out_size, void* d_ws, size_t ws_size,
                              hipStream_t stream) {
    (void)in_sizes; (void)n_in; (void)out_size; (void)ws_size;
    const float* img1 = (const float*)d_in[0];
    const float* img2 = (const float*)d_in[1];
    float* out     = (float*)d_out;
    float* partial = (float*)d_ws;                 // 8192 floats = 32 KB scratch

    const int nblocks = (B_IMG * 16 * 16) / WAVES; // 8192 blocks, 4 tiles each
    ssim_main<<<nblocks, WAVES * 32, 0, stream>>>(img1, img2, partial);
    ssim_finalize<<<1, 256, 0, stream>>>(partial, nblocks, out);
}